// LocalAttentionLayer_79293686218842
// MI455X (gfx1250) — compile-verified
//
#include <hip/hip_runtime.h>
#include <math.h>

// ---------------------------------------------------------------------------
// LocalAttentionLayer for MI455X (gfx1250, wave32).
//   Phase 1: Q = x@Wq+bq, K = ctx@Wk, V = ctx@Wv      (fp32 WMMA 16x16x4)
//   Phase 2: gathered 32-neighbor attention per query  (L2-resident gather)
//   Phase 3: out = O_pre@Wout + bout                   (fp32 WMMA 16x16x4)
//
// All GEMM shapes are compile-time constants so every global access inside
// the K-loop folds into an immediate-offset global_load off one base pointer.
// ---------------------------------------------------------------------------

typedef float v2f __attribute__((ext_vector_type(2)));
typedef float v8f __attribute__((ext_vector_type(8)));

#define HEADS    8
#define HEAD_DIM 64
#define DIM      512
#define D_IN     256
#define BATCH    4
#define SEQ_N    2048
#define SEQ_M    2048
#define KNBR     32

// ---------------------------------------------------------------------------
// fp32 GEMM:  C[M x N] = A[M x Kd] * W[Kd x N] (+ bias[N])
// blockDim = 256 (8 waves). Each wave computes one 16(M) x 32(N) tile:
// two V_WMMA_F32_16X16X4_F32 issues per K-step sharing one A fragment.
//
// Fragment layouts (ISA 7.12.2, wave32):
//   A (16x4):  lanes 0-15 hold rows M=0..15 with {VGPR0=K0, VGPR1=K1};
//              lanes 16-31 hold the same rows with {K2, K3}.
//   B (4x16):  lanes 0-15 hold cols N=0..15 with {VGPR0=K0, VGPR1=K1};
//              lanes 16-31 same cols with {K2, K3}.
//   C/D:       VGPR r, lanes 0-15 -> (M = r,   N = lane),
//                      lanes 16-31 -> (M = r+8, N = lane-16).
// ---------------------------------------------------------------------------
template <int Kd, int Ncols, bool HAS_BIAS>
__global__ __launch_bounds__(256) void gemm_wmma_f32(
    const float* __restrict__ A, const float* __restrict__ W,
    const float* __restrict__ bias, float* __restrict__ C)
{
    const int lane = threadIdx.x & 31;
    const int wave = threadIdx.x >> 5;
    const int l15  = lane & 15;
    const int hi   = lane >> 4;                    // half-wave select (K pair)
    const int n0   = blockIdx.x * 32;              // two 16-wide N tiles
    const int m0   = (blockIdx.y * 8 + wave) * 16;

    // Base pointers; all loop accesses are immediate offsets from these.
    const float* __restrict__ ap = A + (long)(m0 + l15) * Kd + 2 * hi;
    const float* __restrict__ wp = W + (long)(2 * hi) * Ncols + n0 + l15;

    v8f acc0 = {};
    v8f acc1 = {};
#pragma unroll 8
    for (int k0 = 0; k0 < Kd; k0 += 4) {
        // A fragment: 2 consecutive K values for this lane's row (8B aligned)
        v2f a = *(const v2f*)(ap + k0);
        // B fragments for the two N tiles (immediate-offset dword loads)
        v2f b0, b1;
        b0.x = wp[(long)k0 * Ncols];
        b0.y = wp[(long)k0 * Ncols + Ncols];
        b1.x = wp[(long)k0 * Ncols + 16];
        b1.y = wp[(long)k0 * Ncols + Ncols + 16];
        acc0 = __builtin_amdgcn_wmma_f32_16x16x4_f32(
            false, a, false, b0, (short)0, acc0, false, false);
        acc1 = __builtin_amdgcn_wmma_f32_16x16x4_f32(
            false, a, false, b1, (short)0, acc1, false, false);
    }

    float bv0 = 0.0f, bv1 = 0.0f;
    if (HAS_BIAS) {
        bv0 = bias[n0 + l15];
        bv1 = bias[n0 + 16 + l15];
    }
    float* __restrict__ cp = C + (long)(m0 + 8 * hi) * Ncols + n0 + l15;
#pragma unroll
    for (int r = 0; r < 8; ++r) {
        cp[(long)r * Ncols]      = acc0[r] + bv0;
        cp[(long)r * Ncols + 16] = acc1[r] + bv1;
    }
}

// ---------------------------------------------------------------------------
// Gathered local attention. One 256-thread workgroup per query (b, n).
//   sim[kk,h]  = (q[h,:] . K[idx[kk], h,:]) * 64^-0.5      (32x8)
//   attn       = softmax over kk
//   O_pre[o]   = sum_kk attn[kk, o/64] * V[idx[kk], o]     (512 outputs)
// V-gather is the dominant traffic: per kk, all 256 threads read a
// contiguous 2KB row (two coalesced 1KB sweeps) -> full-cacheline L2 reads.
// ---------------------------------------------------------------------------
__global__ __launch_bounds__(256) void local_attn(
    const float* __restrict__ Q, const float* __restrict__ Kp,
    const float* __restrict__ Vp, const int* __restrict__ idxp,
    const unsigned char* __restrict__ mask_q,
    const unsigned char* __restrict__ mask_k,
    float* __restrict__ Opre)
{
    __shared__ int   idx_s[KNBR];
    __shared__ float q_s[DIM];
    __shared__ float sim_s[KNBR * HEADS];

    const int  t    = threadIdx.x;
    const int  n    = blockIdx.x;
    const int  b    = blockIdx.y;
    const long qoff = ((long)(b * SEQ_N + n)) * DIM;

    if (t < KNBR) idx_s[t] = idxp[((long)(b * SEQ_N + n)) * KNBR + t];
    q_s[t]       = Q[qoff + t];
    q_s[t + 256] = Q[qoff + 256 + t];
    __syncthreads();

    // ---- sim = q . k_gathered, masked -------------------------------------
    {
        const int kk = t >> 3;
        const int h  = t & 7;
        const float* __restrict__ krow =
            Kp + ((long)(b * SEQ_M + idx_s[kk])) * DIM + h * HEAD_DIM;
        const float* qh = q_s + h * HEAD_DIM;
        float s = 0.0f;
#pragma unroll
        for (int d = 0; d < HEAD_DIM; d += 4) {
            float4 kv = *(const float4*)(krow + d);
            s += qh[d + 0] * kv.x + qh[d + 1] * kv.y +
                 qh[d + 2] * kv.z + qh[d + 3] * kv.w;
        }
        s *= 0.125f;  // HEAD_DIM^-0.5
        const float BIG_NEG = -3.402823466e+38f;  // finfo(f32).min
        if (!mask_q[b * SEQ_N + n] || !mask_k[b * SEQ_M + idx_s[kk]])
            s = BIG_NEG;
        sim_s[kk * HEADS + h] = s;
    }
    __syncthreads();

    // ---- softmax over the 32 neighbors, one thread per head ---------------
    if (t < HEADS) {
        const int h = t;
        float mx = sim_s[h];
        for (int kk = 1; kk < KNBR; ++kk)
            mx = fmaxf(mx, sim_s[kk * HEADS + h]);
        float sum = 0.0f;
        for (int kk = 0; kk < KNBR; ++kk)
            sum += __expf(sim_s[kk * HEADS + h] - mx);
        const float inv = 1.0f / sum;
        for (int kk = 0; kk < KNBR; ++kk)
            sim_s[kk * HEADS + h] = __expf(sim_s[kk * HEADS + h] - mx) * inv;
    }
    __syncthreads();

    // ---- out = attn-weighted V gather (coalesced 2KB rows) ----------------
    {
        const int h0 = t >> 6;          // head for element t        (0..3)
        const int h1 = (t + 256) >> 6;  // head for element t+256    (4..7)
        float acc0 = 0.0f, acc1 = 0.0f;
        for (int kk = 0; kk < KNBR; ++kk) {
            const float* __restrict__ vrow =
                Vp + ((long)(b * SEQ_M + idx_s[kk])) * DIM;
            acc0 += sim_s[kk * HEADS + h0] * vrow[t];
            acc1 += sim_s[kk * HEADS + h1] * vrow[t + 256];
        }
        Opre[qoff + t]       = acc0;
        Opre[qoff + t + 256] = acc1;
    }
}

// ---------------------------------------------------------------------------
extern "C" void kernel_launch(void* const* d_in, const int* in_sizes, int n_in,
                              void* d_out, int out_size, void* d_ws, size_t ws_size,
                              hipStream_t stream)
{
    const float*         x       = (const float*)d_in[0];
    const float*         context = (const float*)d_in[1];
    const int*           idxp    = (const int*)d_in[2];
    const unsigned char* mq      = (const unsigned char*)d_in[3];
    const unsigned char* mk      = (const unsigned char*)d_in[4];
    const float*         Wq      = (const float*)d_in[5];
    const float*         bq      = (const float*)d_in[6];
    const float*         Wk      = (const float*)d_in[7];
    const float*         Wv      = (const float*)d_in[8];
    const float*         Wout    = (const float*)d_in[9];
    const float*         bout    = (const float*)d_in[10];
    float*               out     = (float*)d_out;

    const long R = (long)BATCH * SEQ_N;      // 8192 rows

    float* ws = (float*)d_ws;                // 4 x 16MB fp32 scratch
    float* Qp = ws;
    float* Kp = Qp + R * DIM;
    float* Vp = Kp + R * DIM;
    float* Op = Vp + R * DIM;

    dim3 blk(256);

    // Projections: [8192 x 256] @ [256 x 512]; 16x32 tile per wave
    dim3 gproj(DIM / 32, (unsigned)(R / 128));
    gemm_wmma_f32<D_IN, DIM, true ><<<gproj, blk, 0, stream>>>(x,       Wq, bq,      Qp);
    gemm_wmma_f32<D_IN, DIM, false><<<gproj, blk, 0, stream>>>(context, Wk, nullptr, Kp);
    gemm_wmma_f32<D_IN, DIM, false><<<gproj, blk, 0, stream>>>(context, Wv, nullptr, Vp);

    // Gathered attention: one workgroup per query
    dim3 gattn(SEQ_N, BATCH);
    local_attn<<<gattn, blk, 0, stream>>>(Qp, Kp, Vp, idxp, mq, mk, Op);

    // Output projection: [8192 x 512] @ [512 x 256] + bout
    dim3 gout(D_IN / 32, (unsigned)(R / 128));
    gemm_wmma_f32<DIM, D_IN, true><<<gout, blk, 0, stream>>>(Op, Wout, bout, out);
}